// MacroGCN_67551245631654
// MI455X (gfx1250) — compile-verified
//
#include <hip/hip_runtime.h>
#include <hip/hip_bf16.h>
#include <math.h>

typedef float v2f __attribute__((ext_vector_type(2)));
typedef float v4f __attribute__((ext_vector_type(4)));
typedef float v8f __attribute__((ext_vector_type(8)));

#define D 64
#define WAVE 32
#define GEMM_TB 256
#define GEMM_WPB 8   // waves per block (GEMM_TB / 32)

// ---------------------------------------------------------------------------
// Kernel 1: x = features @ W via V_WMMA_F32_16X16X4_F32, full tiles only.
// launch_bounds(256,1): allow a big VGPR budget so all 64 B-fragments
// (128 VGPRs) stay register-resident. Pointer-increment grid-stride loop,
// no guards, no clamps, no divergence.
// ---------------------------------------------------------------------------
__global__ void __launch_bounds__(GEMM_TB, 1)
gcn_gemm_wmma(const float* __restrict__ A,   // [N, 64]
              const float* __restrict__ W,   // [64, 64]
              float* __restrict__ X,         // [N, 64]
              int fullTiles) {
  const int wave  = threadIdx.x >> 5;
  const int lane  = threadIdx.x & 31;
  const int mRow  = lane & 15;          // M for A-frag, N-col for B/C frags
  const int kHalf = (lane >> 4) * 2;    // lane group 0 -> K{0,1}; group 1 -> K{2,3}

  // Preload all B fragments: 16 k-steps x 4 N-tiles x v2f = 128 VGPRs.
  v2f bfrag[16][4];
#pragma unroll
  for (int kk = 0; kk < 16; ++kk) {
#pragma unroll
    for (int t = 0; t < 4; ++t) {
      const int k = kk * 4 + kHalf;
      bfrag[kk][t].x = W[(k + 0) * D + t * 16 + mRow];
      bfrag[kk][t].y = W[(k + 1) * D + t * 16 + mRow];
    }
  }

  int tile = blockIdx.x * GEMM_WPB + wave;
  const int stride = gridDim.x * GEMM_WPB;
  const size_t step = (size_t)stride * 16 * D;

  // kHalf baked into the A pointer; row-half offset baked into the X pointer.
  const float* arow = A + (size_t)(tile * 16 + mRow) * D + kHalf;
  float*       xout = X + (size_t)(tile * 16 + ((lane >> 4) * 8)) * D + mRow;

  for (; tile < fullTiles; tile += stride, arow += step, xout += step) {
    v8f acc[4];
#pragma unroll
    for (int t = 0; t < 4; ++t) acc[t] = (v8f){0.f,0.f,0.f,0.f,0.f,0.f,0.f,0.f};

#pragma unroll
    for (int kk = 0; kk < 16; ++kk) {
      v2f a = *(const v2f*)(arow + kk * 4);           // global_load_b64
#pragma unroll
      for (int t = 0; t < 4; ++t) {
        acc[t] = __builtin_amdgcn_wmma_f32_16x16x4_f32(
            false, a, false, bfrag[kk][t], (short)0, acc[t], false, false);
      }
    }

    // C/D layout: VGPR r holds M = r (lanes 0-15) / M = r+8 (lanes 16-31)
#pragma unroll
    for (int t = 0; t < 4; ++t) {
#pragma unroll
      for (int r = 0; r < 8; ++r) {
        xout[(size_t)r * D + t * 16] = acc[t][r];     // unconditional stores
      }
    }
  }
}

// Tail rows (N % 16): plain dot-product, one row per block of 64 threads.
// Launches 0 blocks when N is a multiple of 16 (true for N=50000).
__global__ void gcn_gemm_tail(const float* __restrict__ A, const float* __restrict__ W,
                              float* __restrict__ X, int startRow, int N) {
  int row = startRow + blockIdx.x;
  if (row >= N) return;
  int col = threadIdx.x;                // blockDim = 64
  const float* arow = A + (size_t)row * D;
  float acc = 0.0f;
#pragma unroll
  for (int k = 0; k < D; ++k) acc = fmaf(arow[k], W[k * D + col], acc);
  X[(size_t)row * D + col] = acc;
}

// ---------------------------------------------------------------------------
// Kernel 2a: deg[n] = 1 (self loop), u32 for deterministic accumulation
// ---------------------------------------------------------------------------
__global__ void deg_init(unsigned int* __restrict__ deg, int N) {
  int i = blockIdx.x * blockDim.x + threadIdx.x;
  if (i < N) deg[i] = 1u;
}

// Kernel 2b: histogram over dst (integer atomics -> bit-deterministic)
__global__ void deg_count(const int* __restrict__ dst, unsigned int* __restrict__ deg, int E) {
  int e = blockIdx.x * blockDim.x + threadIdx.x;
  if (e < E) atomicAdd(&deg[dst[e]], 1u);
}

// Kernel 2c: in-place deg(u32) -> dinv(f32) = rsqrt(deg)
__global__ void deg_rsqrt(unsigned int* __restrict__ degbits, int N) {
  int i = blockIdx.x * blockDim.x + threadIdx.x;
  if (i < N) {
    float d = (float)degbits[i];          // deg >= 1 always (self loop)
    ((float*)degbits)[i] = rsqrtf(d);
  }
}

// ---------------------------------------------------------------------------
// Kernel 3: agg init with self-loop contribution: agg[n] = x[n] * dinv[n]^2
// float4-vectorized.
// ---------------------------------------------------------------------------
__global__ void agg_init(const float* __restrict__ X, const float* __restrict__ dinv,
                         float* __restrict__ agg, int totalV4) {
  int i = blockIdx.x * blockDim.x + threadIdx.x;
  if (i < totalV4) {
    int n = i >> 4;                       // 16 float4 per node row
    float di = dinv[n];
    float s = di * di;
    v4f v = ((const v4f*)X)[i];
    v.x *= s; v.y *= s; v.z *= s; v.w *= s;
    ((v4f*)agg)[i] = v;
  }
}

// ---------------------------------------------------------------------------
// Kernel 4: edge scatter. One wave32 per edge; lane owns channels 2L, 2L+1.
// Edge indices + dinv forced uniform -> scalar SMEM loads.
// agg[dst] += x[src] * dinv[src]*dinv[dst]
// ---------------------------------------------------------------------------
__global__ void edge_scatter(const int* __restrict__ src, const int* __restrict__ dst,
                             const float* __restrict__ X, const float* __restrict__ dinv,
                             float* __restrict__ agg, int E) {
  int gid  = blockIdx.x * blockDim.x + threadIdx.x;
  int wave = __builtin_amdgcn_readfirstlane(gid >> 5);   // uniform per wave
  int lane = threadIdx.x & 31;
  if (wave >= E) return;
  int s = src[wave];                                     // scalar loads
  int d = dst[wave];
  float norm = dinv[s] * dinv[d];
  v2f v = *(const v2f*)(X + (size_t)s * D + 2 * lane);   // global_load_b64
  float* ad = agg + (size_t)d * D + 2 * lane;
  atomicAdd(&ad[0], v.x * norm);
  atomicAdd(&ad[1], v.y * norm);
}

// ---------------------------------------------------------------------------
// Kernel 5: per-node fused relu + per-node prediction scalars.
// h = relu(agg + b); s1[n] = h . predW[0:64]; s2[n] = h . predW[64:128]
// One wave32 per node; wave32 xor-shuffle reduction.
// ---------------------------------------------------------------------------
__global__ void node_score(const float* __restrict__ agg, const float* __restrict__ bvec,
                           const float* __restrict__ predW,
                           float* __restrict__ s1, float* __restrict__ s2, int N) {
  int gid  = blockIdx.x * blockDim.x + threadIdx.x;
  int node = __builtin_amdgcn_readfirstlane(gid >> 5);   // uniform per wave
  int lane = threadIdx.x & 31;
  if (node >= N) return;
  const float* an = agg + (size_t)node * D;
  float h0 = fmaxf(an[lane]        + bvec[lane],        0.0f);
  float h1 = fmaxf(an[lane + WAVE] + bvec[lane + WAVE], 0.0f);
  float a = h0 * predW[lane]      + h1 * predW[lane + WAVE];
  float c = h0 * predW[64 + lane] + h1 * predW[96 + lane];
#pragma unroll
  for (int off = 16; off > 0; off >>= 1) {
    a += __shfl_xor(a, off, WAVE);
    c += __shfl_xor(c, off, WAVE);
  }
  if (lane == 0) {
    s1[node] = a;
    s2[node] = c;
  }
}

// ---------------------------------------------------------------------------
// Kernel 6: per-edge logit + sigmoid: out[e] = sigmoid(s1[src]+s2[dst]+predb)
// ---------------------------------------------------------------------------
__global__ void edge_logit(const int* __restrict__ src, const int* __restrict__ dst,
                           const float* __restrict__ s1, const float* __restrict__ s2,
                           const float* __restrict__ predb, float* __restrict__ out, int E) {
  int e = blockIdx.x * blockDim.x + threadIdx.x;
  if (e < E) {
    float z = s1[src[e]] + s2[dst[e]] + predb[0];
    out[e] = 1.0f / (1.0f + __expf(-z));
  }
}

// ---------------------------------------------------------------------------
extern "C" void kernel_launch(void* const* d_in, const int* in_sizes, int n_in,
                              void* d_out, int out_size, void* d_ws, size_t ws_size,
                              hipStream_t stream) {
  const float* features = (const float*)d_in[0];   // [N, 64]
  const int*   edges    = (const int*)d_in[1];     // [2, E]
  const float* W        = (const float*)d_in[2];   // [64, 64]
  const float* bvec     = (const float*)d_in[3];   // [64]
  const float* predW    = (const float*)d_in[4];   // [128, 1]
  const float* predb    = (const float*)d_in[5];   // [1]
  float* out = (float*)d_out;

  const int N = in_sizes[0] / D;
  const int E = in_sizes[1] / 2;
  const int* src = edges;
  const int* dst = edges + E;

  // workspace carve-up
  char* ws = (char*)d_ws;
  float* X   = (float*)ws;                           ws += (size_t)N * D * sizeof(float);
  float* agg = (float*)ws;                           ws += (size_t)N * D * sizeof(float);
  unsigned int* deg = (unsigned int*)ws;             ws += (size_t)N * sizeof(unsigned int);
  float* s1  = (float*)ws;                           ws += (size_t)N * sizeof(float);
  float* s2  = (float*)ws;
  float* dinv = (float*)deg;                         // deg converted in-place to rsqrt

  const int TB = 256;
  const int fullTiles = N / 16;
  const int tailRows  = N - fullTiles * 16;

  // 1. x = features @ W  (WMMA full tiles + scalar tail)
  gcn_gemm_wmma<<<128, GEMM_TB, 0, stream>>>(features, W, X, fullTiles);
  if (tailRows > 0)
    gcn_gemm_tail<<<tailRows, 64, 0, stream>>>(features, W, X, fullTiles * 16, N);
  // 2. degree (deterministic u32) -> dinv
  deg_init <<<(N + TB - 1) / TB, TB, 0, stream>>>(deg, N);
  deg_count<<<(E + TB - 1) / TB, TB, 0, stream>>>(dst, deg, E);
  deg_rsqrt<<<(N + TB - 1) / TB, TB, 0, stream>>>(deg, N);
  // 3. agg = x * dinv^2  (self loops), float4
  {
    int totalV4 = N * (D / 4);
    agg_init<<<(totalV4 + TB - 1) / TB, TB, 0, stream>>>(X, dinv, agg, totalV4);
  }
  // 4. agg[dst] += x[src]*norm  (one wave per edge)
  {
    long long threads = (long long)E * WAVE;
    edge_scatter<<<(unsigned)((threads + TB - 1) / TB), TB, 0, stream>>>(src, dst, X, dinv, agg, E);
  }
  // 5. per-node relu + prediction scalars
  {
    long long threads = (long long)N * WAVE;
    node_score<<<(unsigned)((threads + TB - 1) / TB), TB, 0, stream>>>(agg, bvec, predW, s1, s2, N);
  }
  // 6. per-edge sigmoid
  edge_logit<<<(E + TB - 1) / TB, TB, 0, stream>>>(src, dst, s1, s2, predb, out, E);
}